// MemNN_39788577030921
// MI455X (gfx1250) — compile-verified
//
#include <hip/hip_runtime.h>
#include <math.h>

typedef __attribute__((ext_vector_type(2))) float v2f;
typedef __attribute__((ext_vector_type(8))) float v8f;

#define BATCH 64
#define SLEN  50
#define SWORDS 64
#define EMBED 128
#define QLEN  16
#define VOCAB 100000
#define NTILES (VOCAB / 16)   // 6250 column tiles of 16
#define LCHUNKS 32
#define CHUNKSZ (VOCAB / LCHUNKS)  // 3125

// ---------------------------------------------------------------------------
// u = A[0][q].sum(axis=1)   : 64 blocks x 128 threads (thread = embed dim)
// ---------------------------------------------------------------------------
__global__ void uinit_kernel(const int* __restrict__ q,
                             const float* __restrict__ A0,
                             float* __restrict__ u) {
    int b = blockIdx.x, d = threadIdx.x;
    const int* qb = q + b * QLEN;
    float acc = 0.0f;
#pragma unroll
    for (int j = 0; j < QLEN; ++j) {
        acc += A0[(size_t)qb[j] * EMBED + d];
    }
    u[b * EMBED + d] = acc;
}

// ---------------------------------------------------------------------------
// Hop part A: one block per (batch, sentence) -> 3200 blocks x 128 threads.
//   c[b,s,:]     = sum_w A_h1[x[b,s,w]] + TC[s,:]          (to cbuf scratch)
//   logits[b,s]  = ( sum_w A_h[x[b,s,w]] * pe[w,:] + TA[s,:] ) . u[b,:]
// This is the dominant-traffic phase (~210 MB/hop of embedding gathers);
// 12,800 waves keep the L2/HBM pipes full.
// ---------------------------------------------------------------------------
__global__ void hop_a_kernel(const int* __restrict__ x,
                             const float* __restrict__ Ah,
                             const float* __restrict__ Ah1,
                             const float* __restrict__ TA,
                             const float* __restrict__ TC,
                             const float* __restrict__ u,
                             float* __restrict__ cbuf,
                             float* __restrict__ logits) {
    __shared__ int   xs[SWORDS];
    __shared__ float red[EMBED];

    int s = blockIdx.x, b = blockIdx.y, d = threadIdx.x;

    if (d < SWORDS) xs[d] = x[((size_t)b * SLEN + s) * SWORDS + d];
    __syncthreads();

    float kf  = (float)(d + 1) * (1.0f / (float)EMBED);
    float m_d = TA[s * EMBED + d];
    float c_d = TC[s * EMBED + d];

#pragma unroll 8
    for (int w = 0; w < SWORDS; ++w) {
        int idx = xs[w];  // uniform across block -> broadcast
        size_t off = (size_t)idx * EMBED + d;
        float jf = (float)(w + 1) * (1.0f / (float)SWORDS);
        float pe = 1.0f - jf - kf * (1.0f - 2.0f * jf);
        m_d = fmaf(Ah[off], pe, m_d);
        c_d += Ah1[off];
    }

    cbuf[((size_t)b * SLEN + s) * EMBED + d] = c_d;

    // logit = m . u  (block tree reduction over 128 dims)
    red[d] = m_d * u[b * EMBED + d];
    __syncthreads();
#pragma unroll
    for (int off = EMBED / 2; off > 0; off >>= 1) {
        if (d < off) red[d] += red[d + off];
        __syncthreads();
    }
    if (d == 0) logits[b * SLEN + s] = red[0];
}

// ---------------------------------------------------------------------------
// Hop part B: softmax over 50 logits, then u += p . c.  64 blocks x 128 thr.
// ---------------------------------------------------------------------------
__global__ void hop_b_kernel(const float* __restrict__ cbuf,
                             const float* __restrict__ logits,
                             float* __restrict__ u) {
    __shared__ float ps[SLEN];
    int b = blockIdx.x, d = threadIdx.x;

    if (d < SLEN) ps[d] = logits[b * SLEN + d];
    __syncthreads();

    if (d == 0) {
        float mx = ps[0];
        for (int s = 1; s < SLEN; ++s) mx = fmaxf(mx, ps[s]);
        float sum = 0.0f;
        for (int s = 0; s < SLEN; ++s) {
            float e = expf(ps[s] - mx);
            ps[s] = e;
            sum += e;
        }
        float inv = 1.0f / sum;
        for (int s = 0; s < SLEN; ++s) ps[s] *= inv;
    }
    __syncthreads();

    float acc = 0.0f;
    for (int s = 0; s < SLEN; ++s)
        acc = fmaf(ps[s], cbuf[((size_t)b * SLEN + s) * EMBED + d], acc);
    u[b * EMBED + d] += acc;
}

// ---------------------------------------------------------------------------
// logits = u @ A3^T  via V_WMMA_F32_16X16X4_F32 (fp32 WMMA, memory-bound op).
// Block = 256 threads = 8 waves; each wave computes a 64(M) x 16(N) tile.
// u (64x128 = 32 KB) staged in LDS.
// ---------------------------------------------------------------------------
__global__ void gemm_wmma_kernel(const float* __restrict__ u,
                                 const float* __restrict__ A3,
                                 float* __restrict__ out) {
    __shared__ float us[BATCH * EMBED];
    int tid = threadIdx.x;
    for (int i = tid; i < BATCH * EMBED; i += 256) us[i] = u[i];
    __syncthreads();

    int wave = tid >> 5;
    int lane = tid & 31;
    int nTile = blockIdx.x * 8 + wave;
    if (nTile >= NTILES) return;  // whole-wave uniform exit (EXEC all-1 for WMMA)

    int row   = lane & 15;          // M row (A) / N column (B)
    int khalf = (lane >> 4) * 2;    // which K pair this lane holds
    int ncol  = nTile * 16 + row;

    const float* Bp  = A3 + (size_t)ncol * EMBED + khalf;
    const float* a0p = &us[(0 * 16 + row) * EMBED + khalf];
    const float* a1p = &us[(1 * 16 + row) * EMBED + khalf];
    const float* a2p = &us[(2 * 16 + row) * EMBED + khalf];
    const float* a3p = &us[(3 * 16 + row) * EMBED + khalf];

    v8f acc0 = {}, acc1 = {}, acc2 = {}, acc3 = {};
#pragma unroll 4
    for (int k0 = 0; k0 < EMBED; k0 += 4) {
        v2f bv = *(const v2f*)(Bp + k0);
        v2f a0 = *(const v2f*)(a0p + k0);
        v2f a1 = *(const v2f*)(a1p + k0);
        v2f a2 = *(const v2f*)(a2p + k0);
        v2f a3 = *(const v2f*)(a3p + k0);
        acc0 = __builtin_amdgcn_wmma_f32_16x16x4_f32(false, a0, false, bv, (short)0, acc0, false, false);
        acc1 = __builtin_amdgcn_wmma_f32_16x16x4_f32(false, a1, false, bv, (short)0, acc1, false, false);
        acc2 = __builtin_amdgcn_wmma_f32_16x16x4_f32(false, a2, false, bv, (short)0, acc2, false, false);
        acc3 = __builtin_amdgcn_wmma_f32_16x16x4_f32(false, a3, false, bv, (short)0, acc3, false, false);
    }

    int mbase = 8 * (lane >> 4);
#pragma unroll
    for (int r = 0; r < 8; ++r) {
        out[(size_t)( 0 + mbase + r) * VOCAB + ncol] = acc0[r];
        out[(size_t)(16 + mbase + r) * VOCAB + ncol] = acc1[r];
        out[(size_t)(32 + mbase + r) * VOCAB + ncol] = acc2[r];
        out[(size_t)(48 + mbase + r) * VOCAB + ncol] = acc3[r];
    }
}

// ---------------------------------------------------------------------------
// log_softmax, parallelized: per-chunk partials -> combine -> subtract.
// ---------------------------------------------------------------------------
__global__ void lsm_partial_kernel(const float* __restrict__ out,
                                   float* __restrict__ pmx,
                                   float* __restrict__ psum) {
    __shared__ float red[256];
    int c = blockIdx.x, b = blockIdx.y, tid = threadIdx.x;
    const float* row = out + (size_t)b * VOCAB + (size_t)c * CHUNKSZ;

    float mx = -INFINITY;
    for (int i = tid; i < CHUNKSZ; i += 256) mx = fmaxf(mx, row[i]);
    red[tid] = mx;
    __syncthreads();
#pragma unroll
    for (int off = 128; off > 0; off >>= 1) {
        if (tid < off) red[tid] = fmaxf(red[tid], red[tid + off]);
        __syncthreads();
    }
    mx = red[0];
    __syncthreads();

    float sm = 0.0f;
    for (int i = tid; i < CHUNKSZ; i += 256) sm += expf(row[i] - mx);
    red[tid] = sm;
    __syncthreads();
#pragma unroll
    for (int off = 128; off > 0; off >>= 1) {
        if (tid < off) red[tid] += red[tid + off];
        __syncthreads();
    }
    if (tid == 0) {
        pmx[b * LCHUNKS + c]  = mx;
        psum[b * LCHUNKS + c] = red[0];
    }
}

__global__ void lsm_combine_kernel(const float* __restrict__ pmx,
                                   const float* __restrict__ psum,
                                   float* __restrict__ lse) {
    int b = threadIdx.x;  // 64 threads, 1 block
    float M = -INFINITY;
#pragma unroll
    for (int c = 0; c < LCHUNKS; ++c) M = fmaxf(M, pmx[b * LCHUNKS + c]);
    float S = 0.0f;
#pragma unroll
    for (int c = 0; c < LCHUNKS; ++c)
        S += psum[b * LCHUNKS + c] * expf(pmx[b * LCHUNKS + c] - M);
    lse[b] = M + logf(S);
}

__global__ void lsm_sub_kernel(float* __restrict__ out,
                               const float* __restrict__ lse) {
    int b = blockIdx.y;
    int i = blockIdx.x * 256 + threadIdx.x;
    if (i < VOCAB) out[(size_t)b * VOCAB + i] -= lse[b];
}

// ---------------------------------------------------------------------------
extern "C" void kernel_launch(void* const* d_in, const int* in_sizes, int n_in,
                              void* d_out, int out_size, void* d_ws, size_t ws_size,
                              hipStream_t stream) {
    const int*   x  = (const int*)d_in[0];    // (64, 50, 64)
    const int*   q  = (const int*)d_in[1];    // (64, 16)
    const float* A  = (const float*)d_in[2];  // (4, 100000, 128)
    const float* TA = (const float*)d_in[3];  // (1, 51, 128)
    const float* TC = (const float*)d_in[4];  // (1, 51, 128)
    float* out = (float*)d_out;               // (64, 100000)

    // Workspace layout in d_ws (~62 KB total):
    float* u      = (float*)d_ws;                 // 64*128
    float* logits = u + BATCH * EMBED;            // 64*50
    float* pmx    = logits + BATCH * SLEN;        // 64*32
    float* psum   = pmx + BATCH * LCHUNKS;        // 64*32
    float* lse    = psum + BATCH * LCHUNKS;       // 64
    // Large c scratch (64*50*128 floats = 1.6 MB) lives in the front of
    // d_out; the final GEMM overwrites all of d_out afterwards (stream order).
    float* cbuf   = out;

    const size_t Asz = (size_t)VOCAB * EMBED;

    uinit_kernel<<<BATCH, EMBED, 0, stream>>>(q, A, u);

    for (int h = 0; h < 3; ++h) {
        hop_a_kernel<<<dim3(SLEN, BATCH), EMBED, 0, stream>>>(
            x, A + (size_t)h * Asz, A + (size_t)(h + 1) * Asz,
            TA, TC, u, cbuf, logits);
        hop_b_kernel<<<BATCH, EMBED, 0, stream>>>(cbuf, logits, u);
    }

    gemm_wmma_kernel<<<(NTILES + 7) / 8, 256, 0, stream>>>(u, A + 3 * Asz, out);

    lsm_partial_kernel<<<dim3(LCHUNKS, BATCH), 256, 0, stream>>>(out, pmx, psum);
    lsm_combine_kernel<<<1, BATCH, 0, stream>>>(pmx, psum, lse);
    lsm_sub_kernel<<<dim3((VOCAB + 255) / 256, BATCH), 256, 0, stream>>>(out, lse);
}